// MoChA_81166291960145
// MI455X (gfx1250) — compile-verified
//
#include <hip/hip_runtime.h>
#include <hip/hip_bf16.h>
#include <cstdint>

typedef __attribute__((ext_vector_type(2))) float v2f;
typedef __attribute__((ext_vector_type(8))) float v8f;

#define B_SZ   64
#define K_SZ   2000
#define D_SZ   512
#define A_SZ   512
#define KPAD   2048
#define CHUNKW 8
#define EPSF   1e-6f
#define NEGINF (-3.4028234663852886e38f)   // float32 min, as in reference
#define SCALEF 22.62741699796952f          // sqrt(512)
#define ZSPLIT 4
#define KSEG   500                         // K_SZ / ZSPLIT

// ---------------------------------------------------------------------------
// Small fp32 WMMA GEMM: C[M,N] = A[M,K] * B + (bias)
// MODE 0: B given row-major [K,N]   (used for  q = query @ Wq)
// MODE 1: B given as W[N,K] row-major, i.e. B[k][n] = W[n,k]  (v = q @ Wk^T)
// One wave (32 lanes) per 16x16 output tile; K stepped by 4 per the
// V_WMMA_F32_16X16X4_F32 shape.  fp32 in, fp32 accumulate -> exact.
// A-frag layout (ISA 7.12.2, 32-bit A 16x4): lane<16 holds row=lane, K={0,1};
// lane>=16 holds row=lane-16, K={2,3}.  B-frag symmetric.  C/D: VGPR j holds
// M = j (lanes 0-15) / j+8 (lanes 16-31), N = lane&15.
// ---------------------------------------------------------------------------
template <int MODE>
__global__ __launch_bounds__(32) void wmma_gemm512(const float* __restrict__ A,
                                                   const float* __restrict__ W,
                                                   const float* __restrict__ bias,
                                                   float* __restrict__ C,
                                                   int M, int N, int Kdim) {
  const int m0   = blockIdx.x * 16;
  const int n0   = blockIdx.y * 16;
  const int lane = threadIdx.x;          // 0..31
  const int l15  = lane & 15;
  const int koff = (lane >> 4) * 2;      // 0 or 2

  v8f acc = {};
  const float* arow = A + (size_t)(m0 + l15) * Kdim;

  for (int k0 = 0; k0 < Kdim; k0 += 4) {
    v2f a;
    a.x = arow[k0 + koff];
    a.y = arow[k0 + koff + 1];

    v2f b;
    if (MODE == 0) {
      b.x = W[(size_t)(k0 + koff)     * N + n0 + l15];
      b.y = W[(size_t)(k0 + koff + 1) * N + n0 + l15];
    } else {
      const float* wrow = W + (size_t)(n0 + l15) * Kdim + k0 + koff;
      b.x = wrow[0];
      b.y = wrow[1];
    }
    acc = __builtin_amdgcn_wmma_f32_16x16x4_f32(false, a, false, b,
                                                (short)0, acc, false, false);
  }

  const int nn = n0 + l15;
  const float bv = bias ? bias[nn] : 0.0f;
  const int mbase = m0 + (lane >> 4) * 8;
#pragma unroll
  for (int j = 0; j < 8; ++j)
    C[(size_t)(mbase + j) * N + nn] = acc[j] + bv;
}

// ---------------------------------------------------------------------------
// c_m[b] = q_m[b,:]·bk_mono ; c_c[b] = q_c[b,:]·bk_chunk
// ---------------------------------------------------------------------------
__global__ void compute_c_kernel(const float* __restrict__ q_m,
                                 const float* __restrict__ q_c,
                                 const float* __restrict__ bkm,
                                 const float* __restrict__ bkc,
                                 float* __restrict__ c_m,
                                 float* __restrict__ c_c) {
  __shared__ float r1[256], r2[256];
  const int b = blockIdx.x, t = threadIdx.x;
  float s1 = 0.f, s2 = 0.f;
  for (int i = t; i < A_SZ; i += 256) {
    s1 += q_m[(size_t)b * A_SZ + i] * bkm[i];
    s2 += q_c[(size_t)b * A_SZ + i] * bkc[i];
  }
  r1[t] = s1; r2[t] = s2;
  __syncthreads();
  for (int off = 128; off; off >>= 1) {
    if (t < off) { r1[t] += r1[t + off]; r2[t] += r2[t + off]; }
    __syncthreads();
  }
  if (t == 0) { c_m[b] = r1[0]; c_c[b] = r2[0]; }
}

// ---------------------------------------------------------------------------
// Energies: e_mono[b,k] = (key·v_m + c_m)/scale + r ; e_c = (key·v_c + c_c)/scale
// Streams key once (262 MB) -> the memory-bound hot loop.  float4 loads,
// wave32 shuffle reduction.  grid (B, ceil(K/64)), block 256 (8 waves, 8 k each)
// ---------------------------------------------------------------------------
__global__ __launch_bounds__(256) void energy_kernel(
    const float* __restrict__ key, const float* __restrict__ v_m,
    const float* __restrict__ v_c, const float* __restrict__ c_m,
    const float* __restrict__ c_c, const int* __restrict__ mask,
    const float* __restrict__ r, float* __restrict__ e_mono,
    float* __restrict__ e_c) {
  __shared__ float4 svm[D_SZ / 4];
  __shared__ float4 svc[D_SZ / 4];
  const int b = blockIdx.x;
  const int t = threadIdx.x;
  if (t < D_SZ / 4) {
    svm[t] = ((const float4*)(v_m + (size_t)b * D_SZ))[t];
    svc[t] = ((const float4*)(v_c + (size_t)b * D_SZ))[t];
  }
  __syncthreads();

  const float cm = c_m[b], cc = c_c[b], rv = r[0];
  const int wave = t >> 5, lane = t & 31;

  for (int kk = 0; kk < 8; ++kk) {
    const int k = blockIdx.y * 64 + wave * 8 + kk;
    if (k >= K_SZ) continue;
    const float4* kp = (const float4*)(key + ((size_t)b * K_SZ + k) * D_SZ);
    float dm = 0.f, dc = 0.f;
#pragma unroll
    for (int j = lane; j < D_SZ / 4; j += 32) {
      float4 kv = kp[j];
      float4 m4 = svm[j];
      float4 c4 = svc[j];
      dm += kv.x * m4.x + kv.y * m4.y + kv.z * m4.z + kv.w * m4.w;
      dc += kv.x * c4.x + kv.y * c4.y + kv.z * c4.z + kv.w * c4.w;
    }
    for (int off = 16; off; off >>= 1) {
      dm += __shfl_xor(dm, off, 32);
      dc += __shfl_xor(dc, off, 32);
    }
    if (lane == 0) {
      const bool mk = mask[(size_t)b * K_SZ + k] != 0;
      e_mono[(size_t)b * K_SZ + k] = mk ? (dm + cm) / SCALEF + rv : NEGINF;
      e_c  [(size_t)b * K_SZ + k] = mk ? (dc + cc) / SCALEF      : NEGINF;
    }
  }
}

// ---------------------------------------------------------------------------
// Block-wide inclusive cumsum over KPAD elements (256 threads x 8 items)
// ---------------------------------------------------------------------------
__device__ __forceinline__ void block_cumsum(float* a, float* part) {
  const int t = threadIdx.x;
  float s = 0.f;
#pragma unroll
  for (int i = 0; i < 8; ++i) { s += a[t * 8 + i]; a[t * 8 + i] = s; }
  part[t] = s;
  __syncthreads();
  for (int off = 1; off < 256; off <<= 1) {
    float v = (t >= off) ? part[t - off] : 0.f;
    __syncthreads();
    part[t] += v;
    __syncthreads();
  }
  const float add = (t > 0) ? part[t - 1] : 0.f;
#pragma unroll
  for (int i = 0; i < 8; ++i) a[t * 8 + i] += add;
  __syncthreads();
}

__device__ __forceinline__ float block_max(const float* a, float* part) {
  const int t = threadIdx.x;
  float m = NEGINF;
#pragma unroll
  for (int i = 0; i < 8; ++i) m = fmaxf(m, a[t * 8 + i]);
  part[t] = m;
  __syncthreads();
  for (int off = 128; off; off >>= 1) {
    if (t < off) part[t] = fmaxf(part[t], part[t + off]);
    __syncthreads();
  }
  const float r = part[0];
  __syncthreads();
  return r;
}

__device__ __forceinline__ float sigmoidf_(float x) {
  if (x >= 0.f) { float z = expf(-x); return 1.f / (1.f + z); }
  float z = expf(x); return z / (1.f + z);
}

// ---------------------------------------------------------------------------
// Per-batch monotonic-chunkwise scan.  One block per batch.
// ---------------------------------------------------------------------------
__global__ __launch_bounds__(256) void mocha_scan_kernel(
    const float* __restrict__ e_mono, const float* __restrict__ e_c_in,
    const float* __restrict__ noise, const float* __restrict__ aw_prev,
    float* __restrict__ alpha_out, float* __restrict__ beta_out) {
  __shared__ float b1[KPAD], b2[KPAD], b3[KPAD], b4[KPAD], part[256];
  const int b = blockIdx.x, t = threadIdx.x;

  // ---- p, log(1-p) ----
  for (int k = t; k < KPAD; k += 256) {
    if (k < K_SZ) {
      const float x = e_mono[(size_t)b * K_SZ + k] + noise[(size_t)b * K_SZ + k];
      const float p = sigmoidf_(x);
      b2[k] = p;
      float omp = fminf(fmaxf(1.0f - p, EPSF), 1.0f);
      b1[k] = logf(omp);
    } else { b1[k] = 0.f; b2[k] = 0.f; }
  }
  __syncthreads();
  block_cumsum(b1, part);                 // b1 = inclusive cs of log(1-p)

  // ---- cumprod_1mp (exclusive), g = aw_prev / clip(cumprod) ----
  for (int k = t; k < KPAD; k += 256) {
    const float ex = (k > 0) ? b1[k - 1] : 0.f;
    const float cp = expf(ex);
    b3[k] = cp;
    float g = 0.f;
    if (k < K_SZ) {
      const float cpc = fminf(fmaxf(cp, EPSF), 1.0f);
      g = aw_prev[(size_t)b * K_SZ + k] / cpc;
    }
    b4[k] = g;
  }
  __syncthreads();
  block_cumsum(b4, part);                 // b4 = cumsum(g)

  // ---- alpha ----
  for (int k = t; k < KPAD; k += 256) {
    const float alpha = (k < K_SZ) ? b2[k] * b3[k] * b4[k] : 0.f;
    if (k < K_SZ) alpha_out[(size_t)b * K_SZ + k] = alpha;
    b1[k] = alpha;                        // keep alpha in b1
  }
  __syncthreads();

  // ---- chunkwise softmax ----
  for (int k = t; k < KPAD; k += 256)
    b2[k] = (k < K_SZ) ? e_c_in[(size_t)b * K_SZ + k] : NEGINF;
  __syncthreads();
  const float mx = block_max(b2, part);
  for (int k = t; k < KPAD; k += 256) {
    float sm = 0.f;
    if (k < K_SZ) sm = fmaxf(expf(b2[k] - mx), 1e-5f);
    b2[k] = sm;
    b4[k] = sm;
  }
  __syncthreads();
  block_cumsum(b4, part);                 // b4 = cumsum(sm_exp)

  // ---- u = alpha / moving_sum(sm_exp, back=7) ----
  for (int k = t; k < KPAD; k += 256) {
    float u = 0.f;
    if (k < K_SZ) {
      const float denom = b4[k] - ((k >= CHUNKW) ? b4[k - CHUNKW] : 0.f);
      u = b1[k] / denom;                  // SHARP == 1
    }
    b3[k] = u;
  }
  __syncthreads();
  block_cumsum(b3, part);                 // b3 = cumsum(u)

  // ---- beta = sm_exp * moving_sum(u, fwd=7) ----
  for (int k = t; k < K_SZ; k += 256) {
    const float hi = b3[min(k + CHUNKW - 1, K_SZ - 1)];
    const float lo = (k > 0) ? b3[k - 1] : 0.f;
    beta_out[(size_t)b * K_SZ + k] = b2[k] * (hi - lo);
  }
}

// ---------------------------------------------------------------------------
// cv partials: grid (B, D/256, ZSPLIT); streams value once (262 MB)
// ---------------------------------------------------------------------------
__global__ __launch_bounds__(256) void context_partial_kernel(
    const float* __restrict__ beta, const float* __restrict__ value,
    float* __restrict__ part) {
  __shared__ float sb[KSEG];
  const int b = blockIdx.x;
  const int d = blockIdx.y * 256 + threadIdx.x;
  const int z = blockIdx.z;
  const int k0 = z * KSEG;
  for (int k = threadIdx.x; k < KSEG; k += 256)
    sb[k] = beta[(size_t)b * K_SZ + k0 + k];
  __syncthreads();

  const float* vp = value + ((size_t)b * K_SZ + k0) * D_SZ + d;
  float acc = 0.f;
  for (int k = 0; k < KSEG; ++k)
    acc += sb[k] * vp[(size_t)k * D_SZ];
  part[((size_t)z * B_SZ + b) * D_SZ + d] = acc;
}

__global__ __launch_bounds__(256) void context_reduce_kernel(
    const float* __restrict__ part, float* __restrict__ cv) {
  const int b = blockIdx.x;
  const int d = blockIdx.y * 256 + threadIdx.x;
  float s = 0.f;
#pragma unroll
  for (int z = 0; z < ZSPLIT; ++z)
    s += part[((size_t)z * B_SZ + b) * D_SZ + d];
  cv[(size_t)b * D_SZ + d] = s;
}

// ---------------------------------------------------------------------------
extern "C" void kernel_launch(void* const* d_in, const int* in_sizes, int n_in,
                              void* d_out, int out_size, void* d_ws, size_t ws_size,
                              hipStream_t stream) {
  const float* key      = (const float*)d_in[0];
  const float* value    = (const float*)d_in[1];
  const float* query    = (const float*)d_in[2];
  const int*   mask     = (const int*)  d_in[3];
  const float* aw_prev  = (const float*)d_in[4];
  const float* noise    = (const float*)d_in[5];
  const float* Wk_mono  = (const float*)d_in[6];
  const float* bk_mono  = (const float*)d_in[7];
  const float* Wq_mono  = (const float*)d_in[8];
  const float* bq_mono  = (const float*)d_in[9];
  const float* r        = (const float*)d_in[10];
  const float* Wk_chunk = (const float*)d_in[11];
  const float* bk_chunk = (const float*)d_in[12];
  const float* Wq_chunk = (const float*)d_in[13];
  const float* bq_chunk = (const float*)d_in[14];

  float* ws     = (float*)d_ws;
  float* q_m    = ws;                       // 64*512
  float* q_c    = ws + 32768;               // 64*512
  float* v_m    = ws + 65536;               // 64*512
  float* v_c    = ws + 98304;               // 64*512
  float* c_m    = ws + 131072;              // 64
  float* c_c    = ws + 131136;              // 64
  float* e_mono = ws + 131200;              // 64*2000
  float* e_c    = ws + 259200;              // 64*2000
  float* beta   = ws + 387200;              // 64*2000
  float* cvpart = ws + 515200;              // 4*64*512

  float* cv_out    = (float*)d_out;         // [B,1,D] first
  float* alpha_out = cv_out + B_SZ * D_SZ;  // then [B,K]

  const dim3 gGemm(B_SZ / 16, A_SZ / 16);   // 4 x 32 tiles, 1 wave each

  // q = query @ Wq + bq   (fp32 WMMA)
  wmma_gemm512<0><<<gGemm, 32, 0, stream>>>(query, Wq_mono,  bq_mono,  q_m,
                                            B_SZ, A_SZ, D_SZ);
  wmma_gemm512<0><<<gGemm, 32, 0, stream>>>(query, Wq_chunk, bq_chunk, q_c,
                                            B_SZ, A_SZ, D_SZ);
  // bias scalars
  compute_c_kernel<<<B_SZ, 256, 0, stream>>>(q_m, q_c, bk_mono, bk_chunk,
                                             c_m, c_c);
  // v = q @ Wk^T   (fp32 WMMA)
  wmma_gemm512<1><<<gGemm, 32, 0, stream>>>(q_m, Wk_mono,  nullptr, v_m,
                                            B_SZ, D_SZ, A_SZ);
  wmma_gemm512<1><<<gGemm, 32, 0, stream>>>(q_c, Wk_chunk, nullptr, v_c,
                                            B_SZ, D_SZ, A_SZ);
  // energies (streams key)
  energy_kernel<<<dim3(B_SZ, (K_SZ + 63) / 64), 256, 0, stream>>>(
      key, v_m, v_c, c_m, c_c, mask, r, e_mono, e_c);
  // monotonic + chunkwise scan -> alpha, beta
  mocha_scan_kernel<<<B_SZ, 256, 0, stream>>>(e_mono, e_c, noise, aw_prev,
                                              alpha_out, beta);
  // context vector (streams value)
  context_partial_kernel<<<dim3(B_SZ, D_SZ / 256, ZSPLIT), 256, 0, stream>>>(
      beta, value, cvpart);
  context_reduce_kernel<<<dim3(B_SZ, D_SZ / 256), 256, 0, stream>>>(cvpart,
                                                                    cv_out);
}